// TransformationEstimation_70875550318865
// MI455X (gfx1250) — compile-verified
//
#include <hip/hip_runtime.h>
#include <hip/hip_bf16.h>
#include <math.h>

#define BATCH 8
#define NPTS 2048
#define TILES_PER_BATCH (NPTS / 16) /* 128 */
#define NCHUNKS 8                   /* K split factor */
#define KCHUNK (NPTS / NCHUNKS)     /* 256 */
#define RECS_PER_BATCH (NCHUNKS * TILES_PER_BATCH) /* 1024 */
#define WAVES_PER_WG 4
#define REC_STRIDE 32 /* floats per record: 2 halves x 3 cols x 4 vals, padded */

typedef __attribute__((ext_vector_type(2))) float v2f;
typedef __attribute__((ext_vector_type(8))) float v8f;

// Load one 8x b64 clause (32 consecutive k-values for this lane's row).
__device__ __forceinline__ void load8(v2f (&buf)[8], const float* __restrict__ p)
{
#pragma unroll
    for (int j = 0; j < 8; ++j) buf[j] = *(const v2f*)(p + 4 * j);
}

// Consume one clause: 8 WMMA steps (K advances by 32).
__device__ __forceinline__ void wmma8(v8f& acc, const v2f (&buf)[8],
                                      const float* ldsP, int kblk, int kof,
                                      int nn, float bmask)
{
#pragma unroll
    for (int j = 0; j < 8; ++j) {
        const int kl = kblk + 4 * j + kof; // chunk-local k
        v2f bf;
        bf.x = bmask * ldsP[kl * 3 + nn];
        bf.y = bmask * ldsP[(kl + 1) * 3 + nn];
        acc = __builtin_amdgcn_wmma_f32_16x16x4_f32(
            false, buf[j], false, bf, (short)0, acc, false, false);
    }
}

// ---------------------------------------------------------------------------
// Kernel 1: fused partial (M @ P) -> per-(tile,kchunk) partials of
//           S0 = sum matched (x) Q and msum = sum matched.
// One wave per (16-row tile, 256-K chunk). V_WMMA_F32_16X16X4_F32.
// True ping-pong double buffering: no register rotation, no full drain.
// ---------------------------------------------------------------------------
__global__ __launch_bounds__(WAVES_PER_WG * 32)
void icp_mm_kernel(const float* __restrict__ M,
                   const float* __restrict__ Ppc,
                   const float* __restrict__ Qpc,
                   float* __restrict__ ws)
{
    __shared__ float ldsP[KCHUNK * 3]; // 3 KB: P rows of this WG's k-chunk

    const int b     = blockIdx.x;
    const int chunk = blockIdx.y;
    const int tid   = threadIdx.x;
    const int lane  = tid & 31;
    const int wave  = tid >> 5;
    const int tile  = blockIdx.z * WAVES_PER_WG + wave;
    const int kbase = chunk * KCHUNK;

    const float* Pb = Ppc + (size_t)b * NPTS * 3;
    const float* Qb = Qpc + (size_t)b * NPTS * 3;
    const float* Mb = M   + (size_t)b * NPTS * NPTS;

    // Stage this chunk's P rows into LDS (coalesced), shared by 4 waves.
    for (int i = tid; i < KCHUNK * 3; i += WAVES_PER_WG * 32)
        ldsP[i] = Pb[kbase * 3 + i];
    __syncthreads();

    const int  r0  = tile * 16;
    const int  n   = lane & 15;        // row for A-frag, output col for B/D
    const bool hi  = lane >= 16;
    const int  kof = hi ? 2 : 0;
    // A-fragment stream: lane covers floats (k+kof, k+kof+1) of its row.
    const float* Aptr = Mb + (size_t)(r0 + n) * NPTS + kbase + kof;
    // B-fragment: only output columns 0..2 are live (P has D=3).
    const int   nn    = (n < 3) ? n : 0;
    const float bmask = (n < 3) ? 1.0f : 0.0f;

    v8f acc = {0.f, 0.f, 0.f, 0.f, 0.f, 0.f, 0.f, 0.f};

    // 2-deep software pipeline with static ping-pong buffers.
    v2f buf0[8], buf1[8];
    load8(buf0, Aptr);
    load8(buf1, Aptr + 32);

    int k = 0;
    for (; k + 64 < KCHUNK; k += 64) {
        wmma8(acc, buf0, ldsP, k, kof, nn, bmask);       // waits only on buf0
        load8(buf0, Aptr + k + 64);                      // refill while buf1 flies
        wmma8(acc, buf1, ldsP, k + 32, kof, nn, bmask);  // waits only on buf1
        load8(buf1, Aptr + k + 96);
    }
    wmma8(acc, buf0, ldsP, k, kof, nn, bmask);
    wmma8(acc, buf1, ldsP, k + 32, kof, nn, bmask);

    // Fold partial matched tile against Q rows -> 12 scalars per half-wave.
    if (n < 3) {
        float s0 = 0.f, s1 = 0.f, s2 = 0.f, ms = 0.f;
        const int rbase = r0 + (hi ? 8 : 0);
#pragma unroll
        for (int v = 0; v < 8; ++v) {
            const float  mv = acc[v];
            const float* q  = Qb + (size_t)(rbase + v) * 3;
            s0 += mv * q[0];
            s1 += mv * q[1];
            s2 += mv * q[2];
            ms += mv;
        }
        float* rec = ws + (((size_t)b * NCHUNKS + chunk) * TILES_PER_BATCH + tile)
                            * REC_STRIDE + (hi ? 16 : 0) + n * 4;
        rec[0] = s0; rec[1] = s1; rec[2] = s2; rec[3] = ms;
    }
}

// ---------------------------------------------------------------------------
// Kernel 2: per-batch finalize. Deterministic tree reductions of the 1024
// records and of the P/Q means, then a scalar 3x3 Jacobi SVD -> R, t.
// ---------------------------------------------------------------------------
__global__ __launch_bounds__(256)
void icp_finalize_kernel(const float* __restrict__ ws,
                         const float* __restrict__ Ppc,
                         const float* __restrict__ Qpc,
                         float* __restrict__ out)
{
    const int b   = blockIdx.x;
    const int tid = threadIdx.x;

    __shared__ float redPQ[256 * 6];  // per-thread P/Q partial sums
    __shared__ float sred[256 * 12];  // per-thread S0/msum partial sums

    const float* Pb = Ppc + (size_t)b * NPTS * 3;
    const float* Qb = Qpc + (size_t)b * NPTS * 3;

    float a0 = 0, a1 = 0, a2 = 0, a3 = 0, a4 = 0, a5 = 0;
    for (int i = tid; i < NPTS; i += 256) {
        a0 += Pb[i * 3 + 0]; a1 += Pb[i * 3 + 1]; a2 += Pb[i * 3 + 2];
        a3 += Qb[i * 3 + 0]; a4 += Qb[i * 3 + 1]; a5 += Qb[i * 3 + 2];
    }
    redPQ[tid * 6 + 0] = a0; redPQ[tid * 6 + 1] = a1; redPQ[tid * 6 + 2] = a2;
    redPQ[tid * 6 + 3] = a3; redPQ[tid * 6 + 4] = a4; redPQ[tid * 6 + 5] = a5;

    {
        float part[12];
#pragma unroll
        for (int j = 0; j < 12; ++j) part[j] = 0.f;
        for (int r = tid; r < RECS_PER_BATCH; r += 256) {
            const float* rec = ws + ((size_t)b * RECS_PER_BATCH + r) * REC_STRIDE;
#pragma unroll
            for (int d = 0; d < 3; ++d)
#pragma unroll
                for (int j = 0; j < 4; ++j)
                    part[d * 4 + j] += rec[d * 4 + j] + rec[16 + d * 4 + j];
        }
#pragma unroll
        for (int j = 0; j < 12; ++j) sred[tid * 12 + j] = part[j];
    }
    __syncthreads();

    for (int off = 128; off > 0; off >>= 1) {
        if (tid < off) {
            for (int j = 0; j < 6; ++j)
                redPQ[tid * 6 + j] += redPQ[(tid + off) * 6 + j];
            for (int j = 0; j < 12; ++j)
                sred[tid * 12 + j] += sred[(tid + off) * 12 + j];
        }
        __syncthreads();
    }

    if (tid == 0) {
        const float invN = 1.0f / (float)NPTS;
        float meanP[3], meanQ[3], msum[3], S[3][3];
        for (int e = 0; e < 3; ++e) {
            meanP[e] = redPQ[e] * invN;
            meanQ[e] = redPQ[3 + e] * invN;
        }
        // S = S0 - msum (x) meanQ   (mean-centering of matched cancels in S)
        for (int d = 0; d < 3; ++d) {
            msum[d] = sred[d * 4 + 3];
            for (int e = 0; e < 3; ++e)
                S[d][e] = sred[d * 4 + e] - msum[d] * meanQ[e];
        }

        // ---- 3x3 SVD via Jacobi eigendecomposition of A = S^T S ----
        float A[3][3];
        for (int i = 0; i < 3; ++i)
            for (int j = 0; j < 3; ++j) {
                float v = 0.f;
                for (int kk = 0; kk < 3; ++kk) v += S[kk][i] * S[kk][j];
                A[i][j] = v;
            }
        float V[3][3] = {{1, 0, 0}, {0, 1, 0}, {0, 0, 1}};
        const int PPi[3] = {0, 0, 1}, QQi[3] = {1, 2, 2};
        for (int sweep = 0; sweep < 16; ++sweep) {
            for (int r = 0; r < 3; ++r) {
                const int p = PPi[r], q = QQi[r];
                const float apq = A[p][q];
                if (fabsf(apq) > 1e-30f) {
                    const float tau = (A[q][q] - A[p][p]) / (2.0f * apq);
                    const float tt  = ((tau >= 0.f) ? 1.f : -1.f) /
                                      (fabsf(tau) + sqrtf(1.f + tau * tau));
                    const float c = 1.0f / sqrtf(1.f + tt * tt);
                    const float s = tt * c;
                    for (int kk = 0; kk < 3; ++kk) {
                        const float akp = A[kk][p], akq = A[kk][q];
                        A[kk][p] = c * akp - s * akq;
                        A[kk][q] = s * akp + c * akq;
                    }
                    for (int kk = 0; kk < 3; ++kk) {
                        const float apk = A[p][kk], aqk = A[q][kk];
                        A[p][kk] = c * apk - s * aqk;
                        A[q][kk] = s * apk + c * aqk;
                    }
                    for (int kk = 0; kk < 3; ++kk) {
                        const float vkp = V[kk][p], vkq = V[kk][q];
                        V[kk][p] = c * vkp - s * vkq;
                        V[kk][q] = s * vkp + c * vkq;
                    }
                }
            }
        }
        // U columns: u_i = S v_i / ||S v_i||  (|| || == sigma_i)
        float U[3][3], len[3];
        for (int i = 0; i < 3; ++i) {
            float u0 = 0, u1 = 0, u2 = 0;
            for (int kk = 0; kk < 3; ++kk) {
                u0 += S[0][kk] * V[kk][i];
                u1 += S[1][kk] * V[kk][i];
                u2 += S[2][kk] * V[kk][i];
            }
            const float l = sqrtf(u0 * u0 + u1 * u1 + u2 * u2);
            len[i] = l;
            if (l > 1e-20f) { u0 /= l; u1 /= l; u2 /= l; }
            U[0][i] = u0; U[1][i] = u1; U[2][i] = u2;
        }
        for (int i = 0; i < 3; ++i)
            if (len[i] <= 1e-20f) { // degenerate column: orthogonal complement
                const int j = (i + 1) % 3, k2 = (i + 2) % 3;
                float cx = U[1][j] * U[2][k2] - U[2][j] * U[1][k2];
                float cy = U[2][j] * U[0][k2] - U[0][j] * U[2][k2];
                float cz = U[0][j] * U[1][k2] - U[1][j] * U[0][k2];
                const float l = sqrtf(cx * cx + cy * cy + cz * cz);
                if (l > 1e-20f) { cx /= l; cy /= l; cz /= l; }
                else { cx = 1.f; cy = 0.f; cz = 0.f; }
                U[0][i] = cx; U[1][i] = cy; U[2][i] = cz;
            }
        // R = V U^T ; t = meanQ - R meanP
        float R[3][3];
        for (int a = 0; a < 3; ++a)
            for (int c2 = 0; c2 < 3; ++c2) {
                float v = 0.f;
                for (int i = 0; i < 3; ++i) v += V[a][i] * U[c2][i];
                R[a][c2] = v;
            }
        float* Rout = out + (size_t)b * 9;
        float* tout = out + BATCH * 9 + (size_t)b * 3;
        for (int a = 0; a < 3; ++a) {
            for (int c2 = 0; c2 < 3; ++c2) Rout[a * 3 + c2] = R[a][c2];
            tout[a] = meanQ[a] - (R[a][0] * meanP[0] + R[a][1] * meanP[1] +
                                  R[a][2] * meanP[2]);
        }
    }
}

extern "C" void kernel_launch(void* const* d_in, const int* in_sizes, int n_in,
                              void* d_out, int out_size, void* d_ws, size_t ws_size,
                              hipStream_t stream) {
    (void)in_sizes; (void)n_in; (void)out_size; (void)ws_size;
    const float* Ppc = (const float*)d_in[0];
    const float* Qpc = (const float*)d_in[1];
    const float* M   = (const float*)d_in[2];
    float* ws  = (float*)d_ws;
    float* out = (float*)d_out;

    // 8 batches x 8 K-chunks x 32 tile-groups = 2048 WGs, 4 waves each
    dim3 grid1(BATCH, NCHUNKS, TILES_PER_BATCH / WAVES_PER_WG);
    icp_mm_kernel<<<grid1, dim3(WAVES_PER_WG * 32), 0, stream>>>(M, Ppc, Qpc, ws);
    icp_finalize_kernel<<<dim3(BATCH), dim3(256), 0, stream>>>(ws, Ppc, Qpc, out);
}